// CausalAttentionNaive_12292196401576
// MI455X (gfx1250) — compile-verified
//
#include <hip/hip_runtime.h>
#include <math.h>

// ---------------------------------------------------------------------------
// CausalAttentionNaive on gfx1250 (MI455X), fp32 end-to-end via
// V_WMMA_F32_16X16X4_F32, with double-buffered GLOBAL_LOAD_ASYNC_TO_LDS_B128
// tile staging (ASYNCcnt) on the NN GEMMs.
//
// Pipeline:
//   1) Q = x@Wq (into d_out as scratch), K = x@Wk, V = x@Wv   (ws)
//   2) S = (Q @ K^T) * 1/sqrt(D)   -- strictly-upper 128x128 blocks skipped
//   3) causal softmax rows of S in place (full-row max/Z cancel vs reference)
//   4) out = P @ V with K-loop bounded at the diagonal block
// ---------------------------------------------------------------------------

typedef __attribute__((ext_vector_type(2))) float v2f;
typedef __attribute__((ext_vector_type(8))) float v8f;
typedef int a4i __attribute__((vector_size(16)));   // builtin's pointee type

#define TM 128          // block tile M
#define TN 128          // block tile N
#define TK 32           // block tile K (8 wmma k-steps of 4)
#define ASTR 36         // LDS row stride for A: 144B rows -> 16B aligned for
                        // async B128; banks 4*(9l mod 16), conflict free
#define BSTR 132        // LDS row stride for B: 528B rows -> 16B aligned
// 8 waves per block, arranged 2 (M) x 4 (N); each wave owns 64x32 = 4x2 tiles.

#ifndef __has_builtin
#define __has_builtin(x) 0
#endif
#if __has_builtin(__builtin_amdgcn_global_load_async_to_lds_b128) && \
    __has_builtin(__builtin_amdgcn_s_wait_asynccnt)
#define HAS_ASYNC_LDS 1
#else
#define HAS_ASYNC_LDS 0
#endif

#if HAS_ASYNC_LDS
// Param 1: a4i AS(1)* (global src). Param 2: a4i AS(3)* (LDS dst).
#define GLB_PTR(p) ((__attribute__((address_space(1))) a4i*)(p))
#define LDS_PTR(p) ((__attribute__((address_space(3))) a4i*)(p))
#endif

// ---- 8 k-steps of 4, 8 WMMAs each, on staged LDS tiles ----
__device__ __forceinline__ void wmma_ksteps(const float* As, const float* Bs,
                                            v8f acc[4][2],
                                            int m0, int n0, int half, int l16)
{
    #pragma unroll
    for (int k = 0; k < TK; k += 4) {
        const int kk = k + 2 * half;
        v2f bf[2];
        #pragma unroll
        for (int j = 0; j < 2; j++) {
            const float* bp = &Bs[kk * BSTR + n0 + j * 16 + l16];
            bf[j].x = bp[0];
            bf[j].y = bp[BSTR];
        }
        #pragma unroll
        for (int i = 0; i < 4; i++) {
            const float* ap = &As[(m0 + i * 16 + l16) * ASTR + kk];
            v2f af; af.x = ap[0]; af.y = ap[1];
            #pragma unroll
            for (int j = 0; j < 2; j++) {
                acc[i][j] = __builtin_amdgcn_wmma_f32_16x16x4_f32(
                    false, af, false, bf[j], (short)0, acc[i][j],
                    false, false);
            }
        }
    }
}

// ---- accumulator store: VGPR r holds rows (r, r+8) split by lane half ----
__device__ __forceinline__ void store_acc(float* __restrict__ C, v8f acc[4][2],
                                          int N, int block_m, int block_n,
                                          int m0, int n0, int half, int l16,
                                          float scale)
{
    #pragma unroll
    for (int i = 0; i < 4; i++) {
        #pragma unroll
        for (int j = 0; j < 2; j++) {
            const int rbase = block_m + m0 + i * 16 + half * 8;
            const int cbase = block_n + n0 + j * 16 + l16;
            #pragma unroll
            for (int r = 0; r < 8; r++)
                C[(size_t)(rbase + r) * N + cbase] = acc[i][j][r] * scale;
        }
    }
}

#if HAS_ASYNC_LDS
// Async-stage one 128x32 A tile and one 32x128 B tile (NN) straight into LDS.
__device__ __forceinline__ void prefetch_nn(const float* A, const float* B,
                                            float* As, float* Bs, int K, int N,
                                            int block_m, int block_n, int kb,
                                            int tid)
{
    #pragma unroll
    for (int i = 0; i < 4; i++) {
        int vec = tid + i * 256;            // 0..1023
        int row = vec >> 3;                 // 8 float4 per A row
        int c4  = (vec & 7) << 2;
        __builtin_amdgcn_global_load_async_to_lds_b128(
            GLB_PTR(&A[(size_t)(block_m + row) * K + kb + c4]),
            LDS_PTR(&As[row * ASTR + c4]), 0, 0);
    }
    #pragma unroll
    for (int i = 0; i < 4; i++) {
        int vec = tid + i * 256;
        int row = vec >> 5;                 // 32 float4 per B row
        int c4  = (vec & 31) << 2;
        __builtin_amdgcn_global_load_async_to_lds_b128(
            GLB_PTR(&B[(size_t)(kb + row) * N + block_n + c4]),
            LDS_PTR(&Bs[row * BSTR + c4]), 0, 0);
    }
}
#endif

// ---------------------------------------------------------------------------
// NN GEMM: C = (A[MxK] @ B[KxN]) * scale.
// mode 0: full K.  mode 2: causal K-limit (K-loop stops at diagonal block).
// Double-buffered async-to-LDS staging when available.
// ---------------------------------------------------------------------------
__global__ void __launch_bounds__(256)
gemm_nn_wmma(const float* __restrict__ A, const float* __restrict__ B,
             float* __restrict__ C, int M, int N, int K, float scale, int mode)
{
    const int bx = blockIdx.x, by = blockIdx.y;
    int kEnd = K;
    if (mode == 2) { int lim = (by + 1) * TM; kEnd = lim < K ? lim : K; }

    __shared__ float As[2 * TM * ASTR];
    __shared__ float Bs[2 * TK * BSTR];

    const int tid  = threadIdx.x;
    const int lane = tid & 31;
    const int w    = tid >> 5;
    const int m0   = (w >> 2) * 64;
    const int n0   = (w & 3) * 32;
    const int half = lane >> 4;
    const int l16  = lane & 15;
    const int block_m = by * TM, block_n = bx * TN;

    const v8f vzero = {0.f,0.f,0.f,0.f,0.f,0.f,0.f,0.f};
    v8f acc[4][2];
    #pragma unroll
    for (int i = 0; i < 4; i++)
        #pragma unroll
        for (int j = 0; j < 2; j++)
            acc[i][j] = vzero;

#if HAS_ASYNC_LDS
    const int nb = kEnd / TK;
    prefetch_nn(A, B, As, Bs, K, N, block_m, block_n, 0, tid);
    for (int it = 0; it < nb; ++it) {
        float* curA = As + (it & 1) * (TM * ASTR);
        float* curB = Bs + (it & 1) * (TK * BSTR);
        __builtin_amdgcn_s_wait_asynccnt(0);   // our tile landed in LDS
        __syncthreads();                        // everyone's tile landed; prev
                                                // buffer fully consumed
        if (it + 1 < nb)
            prefetch_nn(A, B,
                        As + ((it + 1) & 1) * (TM * ASTR),
                        Bs + ((it + 1) & 1) * (TK * BSTR),
                        K, N, block_m, block_n, (it + 1) * TK, tid);
        wmma_ksteps(curA, curB, acc, m0, n0, half, l16);
    }
#else
    for (int kb = 0; kb < kEnd; kb += TK) {
        #pragma unroll
        for (int i = 0; i < 4; i++) {
            int vec = tid + i * 256;
            int row = vec >> 3;
            int c4  = (vec & 7) << 2;
            const float4 v = *reinterpret_cast<const float4*>(
                &A[(size_t)(block_m + row) * K + kb + c4]);
            float* d = &As[row * ASTR + c4];
            d[0] = v.x; d[1] = v.y; d[2] = v.z; d[3] = v.w;
        }
        #pragma unroll
        for (int i = 0; i < 4; i++) {
            int vec = tid + i * 256;
            int row = vec >> 5;
            int c4  = (vec & 31) << 2;
            const float4 v = *reinterpret_cast<const float4*>(
                &B[(size_t)(kb + row) * N + block_n + c4]);
            float* d = &Bs[row * BSTR + c4];
            d[0] = v.x; d[1] = v.y; d[2] = v.z; d[3] = v.w;
        }
        __syncthreads();
        wmma_ksteps(As, Bs, acc, m0, n0, half, l16);
        __syncthreads();
    }
#endif

    store_acc(C, acc, N, block_m, block_n, m0, n0, half, l16, scale);
}

// ---------------------------------------------------------------------------
// NT GEMM (scores): C = (A[MxK] @ B[NxK]^T) * scale, strictly-upper 128x128
// blocks skipped (softmax never reads them). Sync staging (transposed scatter
// of B cannot use contiguous async B128).
// ---------------------------------------------------------------------------
__global__ void __launch_bounds__(256)
gemm_nt_wmma_causal(const float* __restrict__ A, const float* __restrict__ B,
                    float* __restrict__ C, int M, int N, int K, float scale)
{
    const int bx = blockIdx.x, by = blockIdx.y;
    if (bx > by) return;

    __shared__ float As[TM * ASTR];
    __shared__ float Bs[TK * BSTR];

    const int tid  = threadIdx.x;
    const int lane = tid & 31;
    const int w    = tid >> 5;
    const int m0   = (w >> 2) * 64;
    const int n0   = (w & 3) * 32;
    const int half = lane >> 4;
    const int l16  = lane & 15;
    const int block_m = by * TM, block_n = bx * TN;

    const v8f vzero = {0.f,0.f,0.f,0.f,0.f,0.f,0.f,0.f};
    v8f acc[4][2];
    #pragma unroll
    for (int i = 0; i < 4; i++)
        #pragma unroll
        for (int j = 0; j < 2; j++)
            acc[i][j] = vzero;

    for (int kb = 0; kb < K; kb += TK) {
        #pragma unroll
        for (int i = 0; i < 4; i++) {
            int vec = tid + i * 256;
            int row = vec >> 3;
            int c4  = (vec & 7) << 2;
            const float4 v = *reinterpret_cast<const float4*>(
                &A[(size_t)(block_m + row) * K + kb + c4]);
            float* d = &As[row * ASTR + c4];
            d[0] = v.x; d[1] = v.y; d[2] = v.z; d[3] = v.w;
        }
        #pragma unroll
        for (int i = 0; i < 4; i++) {
            int vec = tid + i * 256;
            int n   = vec >> 3;
            int k4  = (vec & 7) << 2;
            const float4 v = *reinterpret_cast<const float4*>(
                &B[(size_t)(block_n + n) * K + kb + k4]);
            Bs[(k4 + 0) * BSTR + n] = v.x;
            Bs[(k4 + 1) * BSTR + n] = v.y;
            Bs[(k4 + 2) * BSTR + n] = v.z;
            Bs[(k4 + 3) * BSTR + n] = v.w;
        }
        __syncthreads();
        wmma_ksteps(As, Bs, acc, m0, n0, half, l16);
        __syncthreads();
    }

    store_acc(C, acc, N, block_m, block_n, m0, n0, half, l16, scale);
}

// ---------------------------------------------------------------------------
// Causal softmax, one 256-thread block per row, in place.
// Reference's full-softmax -> tril -> renormalize == causal softmax exactly
// (row max and full-row Z cancel in the ratio). Zeros are written only up to
// the end of the diagonal 128-block, which is all the P@V GEMM will read.
// ---------------------------------------------------------------------------
__global__ void __launch_bounds__(256)
causal_softmax_rows(float* __restrict__ Sbuf, int S)
{
    const int i = blockIdx.x;
    const int t = threadIdx.x;
    float* row = Sbuf + (size_t)i * S;
    const int n = i + 1;                  // valid entries j <= i

    float vals[16];                       // ceil(4096/256) = 16 max
    int   cnt  = 0;
    float lmax = -INFINITY;
    for (int j = t; j < n; j += 256) {
        float v = row[j];
        vals[cnt++] = v;
        lmax = fmaxf(lmax, v);
    }

    __shared__ float red[256];
    red[t] = lmax;
    __syncthreads();
    for (int s = 128; s > 0; s >>= 1) {
        if (t < s) red[t] = fmaxf(red[t], red[t + s]);
        __syncthreads();
    }
    const float m = red[0];
    __syncthreads();

    float lsum = 0.f;
    for (int c = 0; c < cnt; c++) {
        vals[c] = __expf(vals[c] - m);
        lsum += vals[c];
    }
    red[t] = lsum;
    __syncthreads();
    for (int s = 128; s > 0; s >>= 1) {
        if (t < s) red[t] += red[t + s];
        __syncthreads();
    }
    const float inv = 1.0f / red[0];

    cnt = 0;
    for (int j = t; j < n; j += 256) row[j] = vals[cnt++] * inv;

    const int jmax = ((i >> 7) + 1) << 7;           // end of diagonal block
    for (int j = n + t; j < jmax; j += 256) row[j] = 0.f;
}

// ---------------------------------------------------------------------------
extern "C" void kernel_launch(void* const* d_in, const int* in_sizes, int n_in,
                              void* d_out, int out_size, void* d_ws, size_t ws_size,
                              hipStream_t stream)
{
    (void)in_sizes; (void)n_in; (void)out_size; (void)ws_size;
    const int S = 4096, D = 2048;

    const float* x  = (const float*)d_in[0];
    const float* Wq = (const float*)d_in[1];
    const float* Wk = (const float*)d_in[2];
    const float* Wv = (const float*)d_in[3];
    float* out = (float*)d_out;

    // Workspace: K (S*D) | V (S*D) | scores (S*S). Q lives in d_out (dead
    // after the scores GEMM, overwritten by the final P@V GEMM).
    float* Kbuf = (float*)d_ws;
    float* Vbuf = Kbuf + (size_t)S * D;
    float* Sbuf = Vbuf + (size_t)S * D;
    float* Qbuf = out;

    const float scl = 1.0f / sqrtf((float)D);

    dim3 blk(256);
    dim3 gProj(D / TN, S / TM);     // 16 x 32
    gemm_nn_wmma<<<gProj, blk, 0, stream>>>(x, Wq, Qbuf, S, D, D, 1.0f, 0);
    gemm_nn_wmma<<<gProj, blk, 0, stream>>>(x, Wk, Kbuf, S, D, D, 1.0f, 0);
    gemm_nn_wmma<<<gProj, blk, 0, stream>>>(x, Wv, Vbuf, S, D, D, 1.0f, 0);

    dim3 gScore(S / TN, S / TM);    // 32 x 32, upper blocks early-exit
    gemm_nt_wmma_causal<<<gScore, blk, 0, stream>>>(Qbuf, Kbuf, Sbuf, S, S, D, scl);

    causal_softmax_rows<<<S, blk, 0, stream>>>(Sbuf, S);

    dim3 gOut(D / TN, S / TM);      // 16 x 32, causal K-limit
    gemm_nn_wmma<<<gOut, blk, 0, stream>>>(Sbuf, Vbuf, out, S, D, S, 1.0f, 2);
}